// RGCNLayerHeteroAsyn_52329881534839
// MI455X (gfx1250) — compile-verified
//
#include <hip/hip_runtime.h>
#include <math.h>

typedef float v2f __attribute__((ext_vector_type(2)));
typedef float v8f __attribute__((ext_vector_type(8)));

#define DIM 128
#define LP  132   // LDS pitch (floats): stride-4 banks across rows -> conflict-free column reads
#define RT  4     // row-tiles per wave (B-fragment reuse factor)
#define BROWS (16 * RT)

// ---------------------------------------------------------------------------
// Wide GEMM: Y[n_rows x 128] = X @ W + bias.
// Block: 64 rows x 128 cols. 8 waves; wave w owns cols [16w,16w+16) and all 4
// row-tiles, so each B fragment (2 global loads) feeds 4 v_wmma ops.
// ---------------------------------------------------------------------------
__global__ __launch_bounds__(256) void gemm_bias_kernel(
    const float* __restrict__ X, const float* __restrict__ W,
    const float* __restrict__ bias, float* __restrict__ Y, int n_rows) {
  __shared__ float As[BROWS * LP];
  const int row0 = blockIdx.x * BROWS;
  const int tid  = threadIdx.x;
  for (int i = tid; i < BROWS * DIM; i += 256) {
    const int r = i >> 7, c = i & (DIM - 1);
    const int g = row0 + r;
    As[r * LP + c] = (g < n_rows) ? X[(long long)g * DIM + c] : 0.0f;
  }
  __syncthreads();
  const int lane = tid & 31;
  const int half = lane >> 4, lrow = lane & 15;
  const int c0   = (tid >> 5) * 16;
  const int col  = c0 + lrow;

  v8f acc[RT];
#pragma unroll
  for (int t = 0; t < RT; ++t) acc[t] = (v8f){};

#pragma unroll
  for (int kk = 0; kk < DIM; kk += 4) {
    const int ka = kk + 2 * half;          // ISA 16x4 f32 layout: lane half selects K pair
    v2f b;
    b.x = W[ka * DIM + col];
    b.y = W[(ka + 1) * DIM + col];
#pragma unroll
    for (int t = 0; t < RT; ++t) {
      v2f a;
      a.x = As[(t * 16 + lrow) * LP + ka];
      a.y = As[(t * 16 + lrow) * LP + ka + 1];
      acc[t] = __builtin_amdgcn_wmma_f32_16x16x4_f32(false, a, false, b,
                                                     (short)0, acc[t], false, false);
    }
  }
  const float bv = bias[col];
#pragma unroll
  for (int t = 0; t < RT; ++t) {
#pragma unroll
    for (int i = 0; i < 8; ++i) {
      const int g = row0 + t * 16 + i + 8 * half;  // C layout: VGPR i -> M=i / M=i+8
      if (g < n_rows) Y[(long long)g * DIM + col] = acc[t][i] + bv;
    }
  }
}

// ---------------------------------------------------------------------------
// One wave per edge: float4 gather of feat[src], coalesced f32 atomic adds to
// acc[dst]; lane 0 bumps the degree counter.
// ---------------------------------------------------------------------------
__global__ __launch_bounds__(256) void scatter_mean_kernel(
    const float* __restrict__ feat, const int* __restrict__ src,
    const int* __restrict__ dst, float* __restrict__ acc,
    float* __restrict__ deg, int n_edges) {
  const long long t = (long long)blockIdx.x * 256 + threadIdx.x;
  const int e = (int)(t >> 5);
  const int lane = (int)(t & 31);
  if (e >= n_edges) return;
  const int s = src[e], d = dst[e];
  const float4 v = ((const float4*)(feat + (long long)s * DIM))[lane];
  float* ap = acc + (long long)d * DIM + lane * 4;
  atomicAdd(ap + 0, v.x);
  atomicAdd(ap + 1, v.y);
  atomicAdd(ap + 2, v.z);
  atomicAdd(ap + 3, v.w);
  if (lane == 0) atomicAdd(deg + d, 1.0f);
}

// ---------------------------------------------------------------------------
// Fused topic update (3 chained WMMA GEMMs per wave):
//   t     = acc_tt/deg_tt + acc_wt/deg_wt
//   trans = (t @ Wa + ba)*a + h_topic*(1-a)
//   out   = tanh(trans @ W_ih + b_ih + ht_prev @ W_hh + b_hh)
// ---------------------------------------------------------------------------
__device__ __forceinline__ v8f wmma_f32_k128(const float* As,
                                             const float* __restrict__ W,
                                             int c0, int lane, v8f acc) {
  const int half = lane >> 4;
  const int lrow = lane & 15;
  const int col  = c0 + lrow;
#pragma unroll
  for (int kk = 0; kk < DIM; kk += 4) {
    const int ka = kk + 2 * half;
    v2f a, b;
    a.x = As[lrow * LP + ka];
    a.y = As[lrow * LP + ka + 1];
    b.x = W[ka * DIM + col];
    b.y = W[(ka + 1) * DIM + col];
    acc = __builtin_amdgcn_wmma_f32_16x16x4_f32(false, a, false, b,
                                                (short)0, acc, false, false);
  }
  return acc;
}

__global__ __launch_bounds__(256) void topic_update_kernel(
    const float* __restrict__ acc_tt, const float* __restrict__ deg_tt,
    const float* __restrict__ acc_wt, const float* __restrict__ deg_wt,
    const float* __restrict__ h_topic, const float* __restrict__ ht_prev,
    const float* __restrict__ Wa, const float* __restrict__ ba,
    const float* __restrict__ W_ih, const float* __restrict__ b_ih,
    const float* __restrict__ W_hh, const float* __restrict__ b_hh,
    const float* __restrict__ skip, float* __restrict__ topic_out, int n_rows) {
  __shared__ float Ts[16 * LP];   // t_topic tile
  __shared__ float Us[16 * LP];   // trans tile
  __shared__ float Hs[16 * LP];   // ht_prev tile
  const int row0 = blockIdx.x * 16;
  const int tid  = threadIdx.x;
  const float alpha = 1.0f / (1.0f + __expf(-skip[0]));
  for (int i = tid; i < 16 * DIM; i += 256) {
    const int r = i >> 7, c = i & (DIM - 1);
    const int g = row0 + r;
    float v = 0.0f, h = 0.0f;
    if (g < n_rows) {
      const float d1 = fmaxf(deg_tt[g], 1.0f);
      const float d2 = fmaxf(deg_wt[g], 1.0f);
      v = acc_tt[(long long)g * DIM + c] / d1 + acc_wt[(long long)g * DIM + c] / d2;
      h = ht_prev[(long long)g * DIM + c];
    }
    Ts[r * LP + c] = v;
    Hs[r * LP + c] = h;
  }
  __syncthreads();
  const int lane = tid & 31;
  const int c0   = (tid >> 5) * 16;
  const int half = lane >> 4, lrow = lane & 15;
  const int col  = c0 + lrow;

  v8f acc = {};
  acc = wmma_f32_k128(Ts, Wa, c0, lane, acc);
  const float bav = ba[col];
#pragma unroll
  for (int i = 0; i < 8; ++i) {
    const int r = i + 8 * half;
    const int g = row0 + r;
    const float hv = (g < n_rows) ? h_topic[(long long)g * DIM + col] : 0.0f;
    Us[r * LP + col] = (acc[i] + bav) * alpha + hv * (1.0f - alpha);
  }
  __syncthreads();

  v8f acc2 = {};
  acc2 = wmma_f32_k128(Us, W_ih, c0, lane, acc2);   // trans @ W_ih
  acc2 = wmma_f32_k128(Hs, W_hh, c0, lane, acc2);   // + ht_prev @ W_hh
  const float bsum = b_ih[col] + b_hh[col];
#pragma unroll
  for (int i = 0; i < 8; ++i) {
    const int g = row0 + i + 8 * half;
    if (g < n_rows) topic_out[(long long)g * DIM + col] = tanhf(acc2[i] + bsum);
  }
}

// ---------------------------------------------------------------------------
// Fused doc update (wide, 64 rows/block):
//   t   = acc_td/deg_td + acc_wd/deg_wd
//   out = (t @ Wa + ba)*a + h_doc*(1-a)
// ---------------------------------------------------------------------------
__global__ __launch_bounds__(256) void doc_update_kernel(
    const float* __restrict__ acc_td, const float* __restrict__ deg_td,
    const float* __restrict__ acc_wd, const float* __restrict__ deg_wd,
    const float* __restrict__ h_doc, const float* __restrict__ Wa,
    const float* __restrict__ ba, const float* __restrict__ skip,
    float* __restrict__ doc_out, int n_rows) {
  __shared__ float Ts[BROWS * LP];
  const int row0 = blockIdx.x * BROWS;
  const int tid  = threadIdx.x;
  const float a2 = 1.0f / (1.0f + __expf(-skip[0]));
  for (int i = tid; i < BROWS * DIM; i += 256) {
    const int r = i >> 7, c = i & (DIM - 1);
    const int g = row0 + r;
    float v = 0.0f;
    if (g < n_rows) {
      const float d1 = fmaxf(deg_td[g], 1.0f);
      const float d2 = fmaxf(deg_wd[g], 1.0f);
      v = acc_td[(long long)g * DIM + c] / d1 + acc_wd[(long long)g * DIM + c] / d2;
    }
    Ts[r * LP + c] = v;
  }
  __syncthreads();
  const int lane = tid & 31;
  const int half = lane >> 4, lrow = lane & 15;
  const int c0   = (tid >> 5) * 16;
  const int col  = c0 + lrow;

  v8f acc[RT];
#pragma unroll
  for (int t = 0; t < RT; ++t) acc[t] = (v8f){};

#pragma unroll
  for (int kk = 0; kk < DIM; kk += 4) {
    const int ka = kk + 2 * half;
    v2f b;
    b.x = Wa[ka * DIM + col];
    b.y = Wa[(ka + 1) * DIM + col];
#pragma unroll
    for (int t = 0; t < RT; ++t) {
      v2f a;
      a.x = Ts[(t * 16 + lrow) * LP + ka];
      a.y = Ts[(t * 16 + lrow) * LP + ka + 1];
      acc[t] = __builtin_amdgcn_wmma_f32_16x16x4_f32(false, a, false, b,
                                                     (short)0, acc[t], false, false);
    }
  }
  const float bav = ba[col];
#pragma unroll
  for (int t = 0; t < RT; ++t) {
#pragma unroll
    for (int i = 0; i < 8; ++i) {
      const int g = row0 + t * 16 + i + 8 * half;
      if (g < n_rows) {
        const float hv = h_doc[(long long)g * DIM + col];
        doc_out[(long long)g * DIM + col] = (acc[t][i] + bav) * a2 + hv * (1.0f - a2);
      }
    }
  }
}

extern "C" void kernel_launch(void* const* d_in, const int* in_sizes, int n_in,
                              void* d_out, int out_size, void* d_ws, size_t ws_size,
                              hipStream_t stream) {
  (void)n_in; (void)out_size; (void)ws_size;
  const float* h_topic = (const float*)d_in[0];
  const float* h_word  = (const float*)d_in[1];
  const float* h_doc   = (const float*)d_in[2];
  const float* ht_prev = (const float*)d_in[3];
  const int* tt_src = (const int*)d_in[4];
  const int* tt_dst = (const int*)d_in[5];
  const int* wt_src = (const int*)d_in[6];
  const int* wt_dst = (const int*)d_in[7];
  const int* td_src = (const int*)d_in[8];
  const int* td_dst = (const int*)d_in[9];
  const int* wd_src = (const int*)d_in[10];
  const int* wd_dst = (const int*)d_in[11];
  const float* W_tt = (const float*)d_in[12]; const float* b_tt = (const float*)d_in[13];
  const float* W_wt = (const float*)d_in[14]; const float* b_wt = (const float*)d_in[15];
  const float* W_td = (const float*)d_in[16]; const float* b_td = (const float*)d_in[17];
  const float* W_wd = (const float*)d_in[18]; const float* b_wd = (const float*)d_in[19];
  const float* Wa_t = (const float*)d_in[20]; const float* ba_t = (const float*)d_in[21];
  const float* Wa_d = (const float*)d_in[22]; const float* ba_d = (const float*)d_in[23];
  const float* W_ih = (const float*)d_in[24]; const float* b_ih = (const float*)d_in[25];
  const float* W_hh = (const float*)d_in[26]; const float* b_hh = (const float*)d_in[27];
  const float* skip_t = (const float*)d_in[28];
  const float* skip_d = (const float*)d_in[29];

  const int NT = in_sizes[0] / DIM;
  const int NW = in_sizes[1] / DIM;
  const int ND = in_sizes[2] / DIM;
  const int E_TT = in_sizes[4], E_WT = in_sizes[6];
  const int E_TD = in_sizes[8], E_WD = in_sizes[10];

  // Workspace layout (floats); acc buffers reused between topic and doc phases.
  float* ws    = (float*)d_ws;
  float* Wh_a  = ws;                              // NT * 128  (Wh_tt -> Wh_td)
  float* Wh_b  = Wh_a + (size_t)NT * DIM;         // NW * 128  (Wh_wt -> Wh_wd)
  float* acc_a = Wh_b + (size_t)NW * DIM;         // ND * 128
  float* acc_b = acc_a + (size_t)ND * DIM;        // ND * 128
  float* deg_a = acc_b + (size_t)ND * DIM;        // ND
  float* deg_b = deg_a + (size_t)ND;              // ND
  const size_t zero_bytes = ((size_t)2 * ND * DIM + 2 * (size_t)ND) * sizeof(float);

  float* topic_out = (float*)d_out;
  float* doc_out   = topic_out + (size_t)NT * DIM;

  const dim3 blk(256);
  const unsigned gT16 = (unsigned)((NT + 15) / 16);
  const unsigned gT64 = (unsigned)((NT + BROWS - 1) / BROWS);
  const unsigned gW64 = (unsigned)((NW + BROWS - 1) / BROWS);
  const unsigned gD64 = (unsigned)((ND + BROWS - 1) / BROWS);

  // ---- topic phase ----
  hipMemsetAsync(acc_a, 0, zero_bytes, stream);
  gemm_bias_kernel<<<gT64, blk, 0, stream>>>(h_topic, W_tt, b_tt, Wh_a, NT);
  gemm_bias_kernel<<<gW64, blk, 0, stream>>>(h_word,  W_wt, b_wt, Wh_b, NW);
  scatter_mean_kernel<<<(unsigned)(((long long)E_TT * 32 + 255) / 256), blk, 0, stream>>>(
      Wh_a, tt_src, tt_dst, acc_a, deg_a, E_TT);
  scatter_mean_kernel<<<(unsigned)(((long long)E_WT * 32 + 255) / 256), blk, 0, stream>>>(
      Wh_b, wt_src, wt_dst, acc_b, deg_b, E_WT);
  topic_update_kernel<<<gT16, blk, 0, stream>>>(acc_a, deg_a, acc_b, deg_b,
      h_topic, ht_prev, Wa_t, ba_t, W_ih, b_ih, W_hh, b_hh, skip_t, topic_out, NT);

  // ---- doc phase (uses fresh topic_out) ----
  hipMemsetAsync(acc_a, 0, zero_bytes, stream);
  gemm_bias_kernel<<<gT64, blk, 0, stream>>>(topic_out, W_td, b_td, Wh_a, NT);
  gemm_bias_kernel<<<gW64, blk, 0, stream>>>(h_word,    W_wd, b_wd, Wh_b, NW);
  scatter_mean_kernel<<<(unsigned)(((long long)E_TD * 32 + 255) / 256), blk, 0, stream>>>(
      Wh_a, td_src, td_dst, acc_a, deg_a, E_TD);
  scatter_mean_kernel<<<(unsigned)(((long long)E_WD * 32 + 255) / 256), blk, 0, stream>>>(
      Wh_b, wd_src, wd_dst, acc_b, deg_b, E_WD);
  doc_update_kernel<<<gD64, blk, 0, stream>>>(acc_a, deg_a, acc_b, deg_b,
      h_doc, Wa_d, ba_d, skip_d, doc_out, ND);
}